// UserEncoder_74844100100286
// MI455X (gfx1250) — compile-verified
//
#include <hip/hip_runtime.h>
#include <math.h>

typedef __attribute__((ext_vector_type(2))) float v2f;
typedef __attribute__((ext_vector_type(8))) float v8f;

#define B_      64
#define NN_     64
#define HL_     50
#define HLP_    64          // HL padded to multiple of 16 for WMMA K
#define IN_DIM_ 384
#define POS_DIM_ 64
#define NEWS_DIM_ 448
#define ATT_DIM_ 256
#define W1_LD_ (2 * NEWS_DIM_)   // 896

// ---------------------------------------------------------------------------
// Kernel 1: build nf (output #2) and zero-padded lf (workspace).
//   nf[b,n,d]  = d<384 ? news_vec[b,n,d] : pos_emb[0, d-384]
//   lfp[b,h,d] = h<50 ? (d<384 ? log_vec[b,h,d] : pos_emb[1+h, d-384]) : 0
// grid = B*64 rows, 256 threads (two 256/192-wide passes over 448 dims).
// ---------------------------------------------------------------------------
__global__ void build_features(const float* __restrict__ log_vec,
                               const float* __restrict__ news_vec,
                               const float* __restrict__ pos_emb,
                               float* __restrict__ nf_out,
                               float* __restrict__ lfp) {
  const int r   = blockIdx.x;      // b*64 + row
  const int row = r & 63;
  const int b   = r >> 6;
  const int t   = threadIdx.x;
  #pragma unroll
  for (int pass = 0; pass < 2; ++pass) {
    const int d = t + pass * 256;
    if (d >= NEWS_DIM_) break;
    const float nv = (d < IN_DIM_) ? news_vec[r * IN_DIM_ + d]
                                   : pos_emb[d - IN_DIM_];          // row 0
    nf_out[r * NEWS_DIM_ + d] = nv;
    float lv = 0.0f;
    if (row < HL_) {
      lv = (d < IN_DIM_) ? log_vec[(b * HL_ + row) * IN_DIM_ + d]
                         : pos_emb[(1 + row) * POS_DIM_ + (d - IN_DIM_)];
    }
    lfp[r * NEWS_DIM_ + d] = lv;
  }
}

// ---------------------------------------------------------------------------
// Kernel 2: C[M=4096, N=256] = A[4096, 448] * W[:, koff:koff+448]^T
// using V_WMMA_F32_16X16X4_F32. 8 waves/block, one 16x16 tile per wave.
// A-frag (16x4 f32): lane<16 -> K = k0,k0+1 ; lane>=16 -> K = k0+2,k0+3.
// B-frag (4x16): same split, element (k,n) = W[n*896 + koff + k] -> float2.
// ---------------------------------------------------------------------------
__global__ void gemm_nt_wmma(const float* __restrict__ A,
                             const float* __restrict__ W,
                             float* __restrict__ C,
                             int koff) {
  const int lane   = threadIdx.x & 31;
  const int wid    = threadIdx.x >> 5;
  const int mt     = blockIdx.x * 8 + wid;   // M tile (16 rows)
  const int nt     = blockIdx.y;             // N tile (16 cols)
  const int rl     = lane & 15;
  const int khalf  = (lane >> 4) << 1;       // 0 or 2

  const float* Arow = A + (mt * 16 + rl) * NEWS_DIM_ + khalf;
  const float* Wrow = W + (nt * 16 + rl) * W1_LD_ + koff + khalf;

  v8f acc = {};
  #pragma unroll 8
  for (int k0 = 0; k0 < NEWS_DIM_; k0 += 4) {
    v2f a  = *(const v2f*)(Arow + k0);
    v2f bb = *(const v2f*)(Wrow + k0);
    acc = __builtin_amdgcn_wmma_f32_16x16x4_f32(
        false, a, false, bb, (short)0, acc, false, false);
  }

  const int rbase = mt * 16 + ((lane >> 4) << 3);   // D: vgpr j = row j / j+8
  const int col   = nt * 16 + rl;
  #pragma unroll
  for (int j = 0; j < 8; ++j)
    C[(rbase + j) * ATT_DIM_ + col] = acc[j];
}

// ---------------------------------------------------------------------------
// Kernel 3: additive-attention scores + masked softmax.
// One block per (b,n). 8 waves cover HL=50 logits; each wave reduces the
// 256-dim tanh dot-product with wave32 shfl_xor. Wave 0 does the softmax
// and writes a zero-padded 64-wide attention row for the WMMA output GEMM.
// ---------------------------------------------------------------------------
__global__ void attn_kernel(const float* __restrict__ pn,
                            const float* __restrict__ pl,
                            const int*   __restrict__ log_mask,
                            const float* __restrict__ b1,
                            const float* __restrict__ W2,
                            const float* __restrict__ b2,
                            float* __restrict__ attn_ws) {
  const int bn = blockIdx.x;          // b*64 + n
  const int b  = bn >> 6;
  const int t  = threadIdx.x;
  const int lane = t & 31;
  const int wid  = t >> 5;

  __shared__ float pn_s[ATT_DIM_];
  __shared__ float b1_s[ATT_DIM_];
  __shared__ float w2_s[ATT_DIM_];
  __shared__ float logits[HLP_];

  pn_s[t] = pn[bn * ATT_DIM_ + t];
  b1_s[t] = b1[t];
  w2_s[t] = W2[t];
  __syncthreads();

  const float b2v = b2[0];
  for (int h = wid; h < HL_; h += 8) {
    const float* plr = pl + (b * HLP_ + h) * ATT_DIM_;
    float acc = 0.0f;
    #pragma unroll
    for (int a = lane; a < ATT_DIM_; a += 32)
      acc += w2_s[a] * tanhf(pn_s[a] + plr[a] + b1_s[a]);
    #pragma unroll
    for (int off = 16; off; off >>= 1)
      acc += __shfl_xor(acc, off, 32);
    if (lane == 0) {
      float lg = acc + b2v;
      if (log_mask[b * HL_ + h] == 0) lg = -1.0e9f;
      logits[h] = lg;
    }
  }
  __syncthreads();

  if (wid == 0) {
    const int h0 = lane, h1 = lane + 32;
    float x0 = (h0 < HL_) ? logits[h0] : -INFINITY;
    float x1 = (h1 < HL_) ? logits[h1] : -INFINITY;
    float m = fmaxf(x0, x1);
    #pragma unroll
    for (int off = 16; off; off >>= 1)
      m = fmaxf(m, __shfl_xor(m, off, 32));
    float e0 = (h0 < HL_) ? __expf(x0 - m) : 0.0f;
    float e1 = (h1 < HL_) ? __expf(x1 - m) : 0.0f;
    float s = e0 + e1;
    #pragma unroll
    for (int off = 16; off; off >>= 1)
      s += __shfl_xor(s, off, 32);
    const float inv = 1.0f / s;
    attn_ws[bn * HLP_ + h0] = e0 * inv;   // zero for padded h
    attn_ws[bn * HLP_ + h1] = e1 * inv;
  }
}

// ---------------------------------------------------------------------------
// Kernel 4: user_log_vecs[b] = attn[b](64x64, K-padded) * lfp[b](64x448),
// fp32 WMMA, 4 waves/block (one M-tile each), grid (batch, N-tile).
// ---------------------------------------------------------------------------
__global__ void gemm_nn_wmma(const float* __restrict__ attn,
                             const float* __restrict__ lfp,
                             float* __restrict__ out) {
  const int b    = blockIdx.x;
  const int nt   = blockIdx.y;          // 0..27
  const int lane = threadIdx.x & 31;
  const int mt   = threadIdx.x >> 5;    // 0..3
  const int rl   = lane & 15;
  const int khalf = (lane >> 4) << 1;

  const float* Ab = attn + b * HLP_ * HLP_;
  const float* Bb = lfp + b * HLP_ * NEWS_DIM_;

  v8f acc = {};
  #pragma unroll
  for (int k0 = 0; k0 < HLP_; k0 += 4) {
    const int k = k0 + khalf;
    v2f a = *(const v2f*)(Ab + (mt * 16 + rl) * HLP_ + k);
    v2f bb;
    bb.x = Bb[k * NEWS_DIM_ + nt * 16 + rl];
    bb.y = Bb[(k + 1) * NEWS_DIM_ + nt * 16 + rl];
    acc = __builtin_amdgcn_wmma_f32_16x16x4_f32(
        false, a, false, bb, (short)0, acc, false, false);
  }

  const int rbase = mt * 16 + ((lane >> 4) << 3);
  const int col   = nt * 16 + rl;
  #pragma unroll
  for (int j = 0; j < 8; ++j)
    out[(b * NN_ + rbase + j) * NEWS_DIM_ + col] = acc[j];
}

// ---------------------------------------------------------------------------
extern "C" void kernel_launch(void* const* d_in, const int* in_sizes, int n_in,
                              void* d_out, int out_size, void* d_ws, size_t ws_size,
                              hipStream_t stream) {
  const float* log_vec  = (const float*)d_in[0];
  const int*   log_mask = (const int*)  d_in[1];
  const float* news_vec = (const float*)d_in[2];
  const float* pos_emb  = (const float*)d_in[3];
  const float* W1       = (const float*)d_in[4];
  const float* b1       = (const float*)d_in[5];
  const float* W2       = (const float*)d_in[6];
  const float* b2       = (const float*)d_in[7];
  (void)in_sizes; (void)n_in; (void)out_size; (void)ws_size;

  float* out      = (float*)d_out;
  float* user_out = out;                                   // (B, NN, 448)
  float* nf_out   = out + B_ * NN_ * NEWS_DIM_;            // (B, NN, 448)

  float* ws   = (float*)d_ws;
  float* lfp  = ws;                                        // B*64*448
  float* pn   = lfp + B_ * HLP_ * NEWS_DIM_;               // B*64*256
  float* pl   = pn  + B_ * NN_  * ATT_DIM_;                // B*64*256
  float* attn = pl  + B_ * HLP_ * ATT_DIM_;                // B*64*64

  // 1) feature build: nf -> output slot 2, padded lf -> workspace
  build_features<<<dim3(B_ * HLP_), dim3(256), 0, stream>>>(
      log_vec, news_vec, pos_emb, nf_out, lfp);

  // 2) pn = nf * Wn^T  (koff = 0)
  gemm_nt_wmma<<<dim3((B_ * NN_) / 128, ATT_DIM_ / 16), dim3(256), 0, stream>>>(
      nf_out, W1, pn, 0);

  // 3) pl = lfp * Wl^T (koff = 448); padded rows produce unused garbage-free zeros
  gemm_nt_wmma<<<dim3((B_ * HLP_) / 128, ATT_DIM_ / 16), dim3(256), 0, stream>>>(
      lfp, W1, pl, NEWS_DIM_);

  // 4) tanh-logits + masked softmax -> zero-padded attn rows
  attn_kernel<<<dim3(B_ * NN_), dim3(256), 0, stream>>>(
      pn, pl, log_mask, b1, W2, b2, attn);

  // 5) user_log_vecs = attn * lfp (per batch), straight into output slot 1
  gemm_nn_wmma<<<dim3(B_, NEWS_DIM_ / 16), dim3(128), 0, stream>>>(
      attn, lfp, user_out);
}